// GCN_88931592831631
// MI455X (gfx1250) — compile-verified
//
#include <hip/hip_runtime.h>

typedef __attribute__((ext_vector_type(2))) float v2f;
typedef __attribute__((ext_vector_type(4))) float v4f;
typedef __attribute__((ext_vector_type(8))) float v8f;

#define NFEAT 512
#define NHID  128

// ---------------------------------------------------------------------------
// C[M x 128] = A[M x K] @ B[K x 128], row-major, fp32, via V_WMMA_F32_16X16X4_F32.
// grid.x = M/16 (16-row strip per block); 256 threads = 8 waves, wave w owns
// the 16-wide N-tile starting at n0 = 16*w. K must be a multiple of 4.
//
// Fragment layouts (CDNA5 ISA 7.12.2, wave32):
//   A 16x4 f32:  VGPR0 = (M=lane%16, K=2*(lane/16)),   VGPR1 = K+1
//   B  4x16 f32: VGPR0 = (K=2*(lane/16), N=lane%16),   VGPR1 = K+1
//   C 16x16 f32: VGPRj = (M = j + 8*(lane/16), N = lane%16)
// ---------------------------------------------------------------------------
__global__ void __launch_bounds__(256) gemm_wmma_f32(const float* __restrict__ A,
                                                     const float* __restrict__ B,
                                                     float* __restrict__ C,
                                                     int M, int K) {
    const int N    = 128;
    const int m0   = blockIdx.x * 16;
    const int wave = threadIdx.x >> 5;
    const int lane = threadIdx.x & 31;
    const int half = lane >> 4;      // 0 or 1
    const int l16  = lane & 15;
    const int n0   = wave * 16;

    v8f acc = {};

    const float* arow = A + (size_t)(m0 + l16) * K + 2 * half;  // K-pair base
    const float* bcol = B + (size_t)(2 * half) * N + n0 + l16;  // K-pair base

    for (int k0 = 0; k0 < K; k0 += 4) {
        v2f a = *(const v2f*)(arow + k0);            // K = k0+2h, k0+2h+1 (8B aligned)
        v2f b;
        b.x = bcol[(size_t)k0 * N];                  // K = k0+2h
        b.y = bcol[(size_t)(k0 + 1) * N];            // K = k0+2h+1
        acc = __builtin_amdgcn_wmma_f32_16x16x4_f32(
            /*neg_a=*/false, a, /*neg_b=*/false, b,
            /*c_mod=*/(short)0, acc, /*reuse_a=*/false, /*reuse_b=*/false);
    }

    float* crow = C + (size_t)(m0 + 8 * half) * N + n0 + l16;
#pragma unroll
    for (int j = 0; j < 8; ++j)
        crow[(size_t)j * N] = acc[j];
}

// ---------------------------------------------------------------------------
// Scatter SpMM: one wave (32 lanes) per edge; each lane owns a float4 of the
// 128 features. Gather H[col], scale by val, hardware-fp32-atomic-add into
// OUT[row]. Both H and OUT are 25.6 MB -> L2 resident on MI455X (192 MB L2).
// ---------------------------------------------------------------------------
__global__ void __launch_bounds__(256) spmm_scatter(const int*   __restrict__ rows,
                                                    const int*   __restrict__ cols,
                                                    const float* __restrict__ vals,
                                                    const float* __restrict__ H,
                                                    float*       __restrict__ OUT,
                                                    int n_edges) {
    const int wid  = (int)((blockIdx.x * 256u + threadIdx.x) >> 5);
    const int lane = threadIdx.x & 31;
    if (wid >= n_edges) return;

    const int   r = rows[wid];
    const int   c = cols[wid];
    const float v = vals[wid];

    v4f m = *(const v4f*)(H + (size_t)c * NHID + lane * 4);
    float* dst = OUT + (size_t)r * NHID + lane * 4;
    unsafeAtomicAdd(dst + 0, v * m.x);
    unsafeAtomicAdd(dst + 1, v * m.y);
    unsafeAtomicAdd(dst + 2, v * m.z);
    unsafeAtomicAdd(dst + 3, v * m.w);
}

// ---------------------------------------------------------------------------
// Elementwise helpers (float4-vectorized; n4 = total_elems/4, 128 % 4 == 0
// so the bias vector index is just (i mod 32)).
// ---------------------------------------------------------------------------
__global__ void __launch_bounds__(256) zero_f32(float* __restrict__ p, int n4) {
    int i = blockIdx.x * 256 + threadIdx.x;
    if (i < n4) ((v4f*)p)[i] = (v4f){0.f, 0.f, 0.f, 0.f};
}

__global__ void __launch_bounds__(256) bias_relu(const float* __restrict__ S,
                                                 const float* __restrict__ b,
                                                 float* __restrict__ X, int n4) {
    int i = blockIdx.x * 256 + threadIdx.x;
    if (i >= n4) return;
    v4f s  = ((const v4f*)S)[i];
    v4f bb = ((const v4f*)b)[i & 31];
    v4f r;
    r.x = fmaxf(s.x + bb.x, 0.f);
    r.y = fmaxf(s.y + bb.y, 0.f);
    r.z = fmaxf(s.z + bb.z, 0.f);
    r.w = fmaxf(s.w + bb.w, 0.f);
    ((v4f*)X)[i] = r;
}

// out = 0.5 * (x1 + (s2_accumulated_in_out) + b2)
__global__ void __launch_bounds__(256) final_combine(const float* __restrict__ X1,
                                                     const float* __restrict__ b2,
                                                     float* __restrict__ OUT, int n4) {
    int i = blockIdx.x * 256 + threadIdx.x;
    if (i >= n4) return;
    v4f s2 = ((const v4f*)OUT)[i];
    v4f x1 = ((const v4f*)X1)[i];
    v4f bb = ((const v4f*)b2)[i & 31];
    v4f r;
    r.x = 0.5f * (x1.x + s2.x + bb.x);
    r.y = 0.5f * (x1.y + s2.y + bb.y);
    r.z = 0.5f * (x1.z + s2.z + bb.z);
    r.w = 0.5f * (x1.w + s2.w + bb.w);
    ((v4f*)OUT)[i] = r;
}

// ---------------------------------------------------------------------------
extern "C" void kernel_launch(void* const* d_in, const int* in_sizes, int n_in,
                              void* d_out, int out_size, void* d_ws, size_t ws_size,
                              hipStream_t stream) {
    const float* x     = (const float*)d_in[0];
    const int*   erows = (const int*)  d_in[1];
    const int*   ecols = (const int*)  d_in[2];
    const float* evals = (const float*)d_in[3];
    const float* W1    = (const float*)d_in[4];
    const float* b1    = (const float*)d_in[5];
    const float* W2    = (const float*)d_in[6];
    const float* b2    = (const float*)d_in[7];

    const int M       = in_sizes[0] / NFEAT;  // 50000 (multiple of 16)
    const int n_edges = in_sizes[1];          // 800000

    float* h  = (float*)d_ws;                 // [M, 128]  (h, then h2)
    float* s1 = h + (size_t)M * NHID;         // [M, 128]  (s1, then x1)
    float* out = (float*)d_out;               // [M, 128]  (s2 accum, then result)

    const int n4 = M * NHID / 4;
    const int eblocks = (n_edges + 7) / 8;    // 8 waves (edges) per 256-thread block
    const int vblocks = (n4 + 255) / 256;

    // 1) h = x @ W1
    gemm_wmma_f32<<<M / 16, 256, 0, stream>>>(x, W1, h, M, NFEAT);

    // 2) s1 = Adj @ h (zero + atomic scatter)
    zero_f32<<<vblocks, 256, 0, stream>>>(s1, n4);
    spmm_scatter<<<eblocks, 256, 0, stream>>>(erows, ecols, evals, h, s1, n_edges);

    // 3) x1 = relu(s1 + b1)   (in place)
    bias_relu<<<vblocks, 256, 0, stream>>>(s1, b1, s1, n4);

    // 4) h2 = x1 @ W2   (reuse h)
    gemm_wmma_f32<<<M / 16, 256, 0, stream>>>(s1, W2, h, M, NHID);

    // 5) s2 = Adj @ h2, accumulated directly into d_out
    zero_f32<<<vblocks, 256, 0, stream>>>(out, n4);
    spmm_scatter<<<eblocks, 256, 0, stream>>>(erows, ecols, evals, h, out, n_edges);

    // 6) out = 0.5 * (x1 + s2 + b2)
    final_combine<<<vblocks, 256, 0, stream>>>(s1, b2, out, n4);
}